// PlanarFlow_61649960567178
// MI455X (gfx1250) — compile-verified
//
#include <hip/hip_runtime.h>

// PlanarFlow on MI455X (gfx1250):
//   Pass 1: per-row triple dots (wu, wz, ww) via async global->LDS streaming
//           (GLOBAL_LOAD_ASYNC_TO_LDS_B128 GVS form + s_wait_asynccnt, 4-deep pipeline)
//   Pass 2: single-block scalar pass: w_norm = sqrt(sum ww), alpha, t, log_det
//   Pass 3: elementwise z_new = z + t*(u + alpha*w); w/u/z re-reads hit the 192MB L2,
//           z_new stored non-temporally so it does not evict the streaming inputs.
// inner2 is computed analytically: inner2 = inner + tanh(inner)*(wu + alpha*ww),
// which avoids a third full pass over memory.

#define PF_BS 8192
#define PF_D  1024

typedef __attribute__((ext_vector_type(4))) float pf_f32x4;

// ---------------------------------------------------------------------------
// CDNA5 async-copy helper (GVS addressing: sgpr base + 32-bit vgpr offset).
// Tracked by ASYNCcnt. The base is wave-uniform; the "s" constraint makes the
// backend place it in an SGPR pair (readfirstlane), eliminating per-chunk
// 64-bit VALU address arithmetic.
// ---------------------------------------------------------------------------
__device__ __forceinline__ void pf_async_b128(unsigned lds_off, unsigned voff,
                                              unsigned long long sbase) {
    asm volatile("global_load_async_to_lds_b128 %0, %1, %2"
                 :: "v"(lds_off), "v"(voff), "s"(sbase)
                 : "memory");
}

// ---------------------------------------------------------------------------
// Kernel 1: per-row dot products wu = w.u, wz = w.z, ww = w.w
// One wave (32 lanes) per row; 8 waves per block; 4-stage async LDS pipeline.
// Each chunk = 128 floats/array (32 lanes x float4). 8 chunks cover D=1024.
// ---------------------------------------------------------------------------
__global__ __launch_bounds__(256) void pf_rowdots(const float* __restrict__ w,
                                                  const float* __restrict__ u,
                                                  const float* __restrict__ z,
                                                  float* __restrict__ ws_wu,
                                                  float* __restrict__ ws_wz,
                                                  float* __restrict__ ws_ww) {
    // 8 waves * 4 buffers * 3 arrays * 128 floats = 48 KB LDS
    __shared__ __align__(16) float smem[8 * 4 * 3 * 128];

    const int wid  = (int)(threadIdx.x >> 5);
    const int lane = (int)(threadIdx.x & 31);
    const int row  = (int)blockIdx.x * 8 + wid;

    const unsigned long long base = (unsigned long long)row * PF_D;
    // Wave-uniform row base addresses (become SGPR pairs inside the asm).
    const unsigned long long wb = (unsigned long long)(size_t)(w + base);
    const unsigned long long ub = (unsigned long long)(size_t)(u + base);
    const unsigned long long zb = (unsigned long long)(size_t)(z + base);

    float* wavebuf = &smem[wid * (4 * 3 * 128)];
    const unsigned lbase = (unsigned)(size_t)wavebuf;
    const unsigned llane = (unsigned)lane * 16u;           // lane*4 floats in bytes

    auto issue = [&](int c) {
        const unsigned bsel = (unsigned)(c & 3);
        const unsigned voff = (unsigned)c * 512u + llane;  // byte offset in row
        const unsigned lw = lbase + bsel * 1536u + 0u   + llane;
        const unsigned lu = lbase + bsel * 1536u + 512u + llane;
        const unsigned lz = lbase + bsel * 1536u + 1024u + llane;
        pf_async_b128(lw, voff, wb);
        pf_async_b128(lu, voff, ub);
        pf_async_b128(lz, voff, zb);
    };

    // Prime the pipeline with chunks 0..2 (9 async ops in flight).
    issue(0); issue(1); issue(2);

    float pwu = 0.0f, pwz = 0.0f, pww = 0.0f;

    #pragma unroll
    for (int c = 0; c < 8; ++c) {
        if (c + 3 < 8) issue(c + 3);
        // Drain chunk c: allow up to 3 pending chunks (9 ops) behind it.
        const int rem = (7 - c) < 3 ? (7 - c) : 3;
        if (rem == 3)      asm volatile("s_wait_asynccnt 0x9" ::: "memory");
        else if (rem == 2) asm volatile("s_wait_asynccnt 0x6" ::: "memory");
        else if (rem == 1) asm volatile("s_wait_asynccnt 0x3" ::: "memory");
        else               asm volatile("s_wait_asynccnt 0x0" ::: "memory");

        const int bsel = c & 3;
        const float4 wv = *(const float4*)&wavebuf[bsel * 384 + 0 * 128 + lane * 4];
        const float4 uv = *(const float4*)&wavebuf[bsel * 384 + 1 * 128 + lane * 4];
        const float4 zv = *(const float4*)&wavebuf[bsel * 384 + 2 * 128 + lane * 4];

        pwu = fmaf(wv.x, uv.x, pwu); pwu = fmaf(wv.y, uv.y, pwu);
        pwu = fmaf(wv.z, uv.z, pwu); pwu = fmaf(wv.w, uv.w, pwu);
        pwz = fmaf(wv.x, zv.x, pwz); pwz = fmaf(wv.y, zv.y, pwz);
        pwz = fmaf(wv.z, zv.z, pwz); pwz = fmaf(wv.w, zv.w, pwz);
        pww = fmaf(wv.x, wv.x, pww); pww = fmaf(wv.y, wv.y, pww);
        pww = fmaf(wv.z, wv.z, pww); pww = fmaf(wv.w, wv.w, pww);
    }

    // wave32 butterfly reduction
    #pragma unroll
    for (int m = 16; m >= 1; m >>= 1) {
        pwu += __shfl_xor(pwu, m, 32);
        pwz += __shfl_xor(pwz, m, 32);
        pww += __shfl_xor(pww, m, 32);
    }

    if (lane == 0) {
        ws_wu[row] = pwu;
        ws_wz[row] = pwz;
        ws_ww[row] = pww;
    }
}

// ---------------------------------------------------------------------------
// Kernel 2: global w_norm reduction + all per-row scalar math (tiny).
// Single block of 1024 threads; deterministic tree reduction.
// ---------------------------------------------------------------------------
__global__ __launch_bounds__(1024) void pf_scalars(const float* __restrict__ b,
                                                   const float* __restrict__ ws_wu,
                                                   const float* __restrict__ ws_wz,
                                                   const float* __restrict__ ws_ww,
                                                   float* __restrict__ ws_alpha,
                                                   float* __restrict__ ws_t,
                                                   float* __restrict__ logdet) {
    __shared__ float red[1024];
    const int t = (int)threadIdx.x;

    float s = 0.0f;
    #pragma unroll
    for (int k = t; k < PF_BS; k += 1024) s += ws_ww[k];
    red[t] = s;
    __syncthreads();
    #pragma unroll
    for (int off = 512; off >= 1; off >>= 1) {
        if (t < off) red[t] += red[t + off];
        __syncthreads();
    }
    const float w_norm = sqrtf(red[0]);

    #pragma unroll
    for (int k = t; k < PF_BS; k += 1024) {
        const float wu = ws_wu[k];
        const float wz = ws_wz[k];
        const float ww = ws_ww[k];
        const float bb = b[k];

        // stable softplus: log(1+e^x) = max(x,0) + log1p(e^{-|x|})
        const float sp    = fmaxf(wu, 0.0f) + log1pf(expf(-fabsf(wu)));
        const float m     = sp - 1.0f;
        const float alpha = (m - wu) / w_norm;

        const float inner = wz + bb;
        const float th    = tanhf(inner);

        const float wuhat  = fmaf(alpha, ww, wu);          // sum(w * u_hat)
        const float inner2 = fmaf(th, wuhat, inner);       // sum(w * z_new) + b
        const float t2     = tanhf(inner2);
        const float hp     = 1.0f - t2 * t2;
        const float ld     = logf(fabsf(fmaf(hp, wuhat, 1.0f)));

        ws_alpha[k] = alpha;
        ws_t[k]     = th;
        logdet[k]   = ld;
    }
}

// ---------------------------------------------------------------------------
// Kernel 3: elementwise z_new = z + t*(u + alpha*w). One block per row,
// 256 threads x float4 = 1024 cols. w/u/z reads hit L2; z_new is stored
// non-temporally (write-only, never re-read) to avoid evicting the inputs.
// ---------------------------------------------------------------------------
__global__ __launch_bounds__(256) void pf_apply(const float* __restrict__ w,
                                                const float* __restrict__ u,
                                                const float* __restrict__ z,
                                                const float* __restrict__ ws_alpha,
                                                const float* __restrict__ ws_t,
                                                float* __restrict__ znew) {
    const int row = (int)blockIdx.x;
    const float a = ws_alpha[row];   // uniform -> scalar load
    const float t = ws_t[row];       // uniform -> scalar load

    const unsigned long long base = (unsigned long long)row * PF_D + (unsigned long long)threadIdx.x * 4;
    const float4 wv = *(const float4*)(w + base);
    const float4 uv = *(const float4*)(u + base);
    const float4 zv = *(const float4*)(z + base);

    pf_f32x4 o;
    o.x = fmaf(t, fmaf(a, wv.x, uv.x), zv.x);
    o.y = fmaf(t, fmaf(a, wv.y, uv.y), zv.y);
    o.z = fmaf(t, fmaf(a, wv.z, uv.z), zv.z);
    o.w = fmaf(t, fmaf(a, wv.w, uv.w), zv.w);
    __builtin_nontemporal_store(o, (pf_f32x4*)(znew + base));
}

// ---------------------------------------------------------------------------
extern "C" void kernel_launch(void* const* d_in, const int* in_sizes, int n_in,
                              void* d_out, int out_size, void* d_ws, size_t ws_size,
                              hipStream_t stream) {
    (void)in_sizes; (void)n_in; (void)out_size; (void)ws_size;

    const float* z = (const float*)d_in[0];
    const float* w = (const float*)d_in[1];
    const float* u = (const float*)d_in[2];
    const float* b = (const float*)d_in[3];

    float* out    = (float*)d_out;
    float* znew   = out;                                  // [8192*1024]
    float* logdet = out + (size_t)PF_BS * PF_D;           // [8192]

    float* wsf      = (float*)d_ws;                       // 5*8192 floats = 160 KB
    float* ws_wu    = wsf + 0 * PF_BS;
    float* ws_wz    = wsf + 1 * PF_BS;
    float* ws_ww    = wsf + 2 * PF_BS;
    float* ws_alpha = wsf + 3 * PF_BS;
    float* ws_t     = wsf + 4 * PF_BS;

    pf_rowdots<<<PF_BS / 8, 256, 0, stream>>>(w, u, z, ws_wu, ws_wz, ws_ww);
    pf_scalars<<<1, 1024, 0, stream>>>(b, ws_wu, ws_wz, ws_ww, ws_alpha, ws_t, logdet);
    pf_apply<<<PF_BS, 256, 0, stream>>>(w, u, z, ws_alpha, ws_t, znew);
}